// MoE_3616362463841
// MI455X (gfx1250) — compile-verified
//
#include <hip/hip_runtime.h>

typedef float v2f __attribute__((ext_vector_type(2)));
typedef float v8f __attribute__((ext_vector_type(8)));

#define B_N 4096
#define E_N 8
#define P_N 256
#define K_N 16
#define F_N 64

// workspace layout (floats)
#define WS_BMAT1 0      // 16x16, [k][n]: n<8 -> gw[n][k], n>=8 -> W0[n-8][k]
#define WS_BMAT2 256    // 16x16, [k][n]: n<8 -> W1[n][k], else 0
#define WS_GBP   512    // 8: P*gb[e]
#define WS_B0    520    // 8: P*sum_{f<F} eb[e][f]
#define WS_B1    528    // 8: P*sum_{f>=F} eb[e][f]
#define WS_PART  536    // 256 blocks * 16 (8 h-sums, 8 mask-counts)

__global__ void moe_prep_kernel(const float* __restrict__ gw,
                                const float* __restrict__ gb,
                                const float* __restrict__ ew,
                                const float* __restrict__ eb,
                                float* __restrict__ ws)
{
    const int tid = threadIdx.x;          // 256 threads
    const int n = tid & 15;
    const int k = tid >> 4;

    float v;
    if (n < E_N) {
        v = gw[n * K_N + k];
    } else {
        const int e = n - E_N;
        float s = 0.f;
        for (int f = 0; f < F_N; ++f) s += ew[(e * 2 * F_N + f) * K_N + k];
        v = s;
    }
    ws[WS_BMAT1 + k * 16 + n] = v;

    if (n < E_N) {
        float s = 0.f;
        for (int f = F_N; f < 2 * F_N; ++f) s += ew[(n * 2 * F_N + f) * K_N + k];
        v = s;
    } else {
        v = 0.f;
    }
    ws[WS_BMAT2 + k * 16 + n] = v;

    if (tid < E_N) {
        ws[WS_GBP + tid] = (float)P_N * gb[tid];
        float s0 = 0.f, s1 = 0.f;
        for (int f = 0; f < F_N; ++f)       s0 += eb[tid * 2 * F_N + f];
        for (int f = F_N; f < 2 * F_N; ++f) s1 += eb[tid * 2 * F_N + f];
        ws[WS_B0 + tid] = (float)P_N * s0;
        ws[WS_B1 + tid] = (float)P_N * s1;
    }
}

__global__ __launch_bounds__(512)
void moe_main_kernel(const float* __restrict__ x,
                     const float* __restrict__ noise,
                     const float* __restrict__ ws,
                     float* __restrict__ part,
                     float* __restrict__ out,
                     float* __restrict__ mask_out)
{
    __shared__ float s_tile[16 * 16];   // s[b_tile][k]
    __shared__ float res1[16 * 16];     // D of WMMA #1: cols 0-7 = h (pre-bias), 8-15 = s.W0
    __shared__ float res2[16 * 16];     // D of WMMA #2: cols 0-7 = s.W1
    __shared__ float hl[8 * 16];        // h[e][m] for loss partials
    __shared__ int   idxs[16];

    const int tid  = threadIdx.x;
    const int wave = tid >> 5;          // 16 waves, one batch row each
    const int lane = tid & 31;
    const int brow = blockIdx.x * 16 + wave;

    // ---- bandwidth phase: s[k] = sum_p x[b][p][k], float4 (b128) loads ----
    const float4* row = (const float4*)(x + (size_t)brow * (P_N * K_N));
    float4 a = make_float4(0.f, 0.f, 0.f, 0.f);
    #pragma unroll
    for (int i = 0; i < 32; ++i) {
        float4 v = row[lane + 32 * i];
        a.x += v.x; a.y += v.y; a.z += v.z; a.w += v.w;
    }
    // lanes with equal (lane&3) hold the same k-quad; reduce across them
    #pragma unroll
    for (int off = 4; off < 32; off <<= 1) {
        a.x += __shfl_xor(a.x, off, 32);
        a.y += __shfl_xor(a.y, off, 32);
        a.z += __shfl_xor(a.z, off, 32);
        a.w += __shfl_xor(a.w, off, 32);
    }
    if (lane < 4) {
        s_tile[wave * 16 + lane * 4 + 0] = a.x;
        s_tile[wave * 16 + lane * 4 + 1] = a.y;
        s_tile[wave * 16 + lane * 4 + 2] = a.z;
        s_tile[wave * 16 + lane * 4 + 3] = a.w;
    }
    __syncthreads();

    // ---- WMMA phase (wave 0): [16x16] s-tile x two 16x16 weight mats ----
    if (tid < 32) {
        const int hi = lane >> 4;       // 0: K pair {0,1}; 1: K pair {2,3}
        const int m  = lane & 15;       // A row / B column
        v8f acc1 = {};
        v8f acc2 = {};
        const float* Bm1 = ws + WS_BMAT1;
        const float* Bm2 = ws + WS_BMAT2;
        #pragma unroll
        for (int ks = 0; ks < 4; ++ks) {
            const int k0 = ks * 4 + hi * 2;
            v2f av; av.x = s_tile[m * 16 + k0];  av.y = s_tile[m * 16 + k0 + 1];
            v2f b1; b1.x = Bm1[k0 * 16 + m];     b1.y = Bm1[(k0 + 1) * 16 + m];
            v2f b2; b2.x = Bm2[k0 * 16 + m];     b2.y = Bm2[(k0 + 1) * 16 + m];
            acc1 = __builtin_amdgcn_wmma_f32_16x16x4_f32(
                false, av, false, b1, (short)0, acc1, false, false);
            acc2 = __builtin_amdgcn_wmma_f32_16x16x4_f32(
                false, av, false, b2, (short)0, acc2, false, false);
        }
        // D layout: VGPR r holds row (r + 8*hi), column = lane&15
        #pragma unroll
        for (int r = 0; r < 8; ++r) {
            const int rowi = r + hi * 8;
            res1[rowi * 16 + m] = acc1[r];
            res2[rowi * 16 + m] = acc2[r];
        }
    }
    __syncthreads();

    // ---- finalize: per-row argmax, output, mask, loss partials ----
    if (tid < 16) {
        const int m = tid;
        const int b = blockIdx.x * 16 + m;
        const float nz = noise[b];
        float hbest = -3.402823466e38f;
        int best = 0;
        #pragma unroll
        for (int e = 0; e < E_N; ++e) {
            const float he = res1[m * 16 + e] + ws[WS_GBP + e] + nz;
            hl[e * 16 + m] = he;
            if (he > hbest) { hbest = he; best = e; }   // strict >: first-max (jnp.argmax)
        }
        idxs[m] = best;
        const float d0 = res1[m * 16 + 8 + best] + ws[WS_B0 + best];
        const float d1 = res2[m * 16 + best]     + ws[WS_B1 + best];
        out[(size_t)b * 2 + 0] = hbest * d0;
        out[(size_t)b * 2 + 1] = hbest * d1;
        #pragma unroll
        for (int e = 0; e < E_N; ++e)
            mask_out[(size_t)b * 8 + e] = (e == best) ? 1.f : 0.f;
    }
    __syncthreads();

    if (tid < 8) {
        float hs = 0.f, cs = 0.f;
        #pragma unroll
        for (int m = 0; m < 16; ++m) {
            hs += hl[tid * 16 + m];
            cs += (idxs[m] == tid) ? 1.f : 0.f;
        }
        part[blockIdx.x * 16 + tid]     = hs;
        part[blockIdx.x * 16 + 8 + tid] = cs;
    }
}

__global__ void moe_loss_kernel(const float* __restrict__ part,
                                float* __restrict__ out_loss)
{
    __shared__ float lh[256];
    __shared__ float lc[256];
    __shared__ float lt[8];
    const int t = threadIdx.x;          // 256 threads
    const int e = t & 7;
    const int g = t >> 3;
    float hs = 0.f, cs = 0.f;
    #pragma unroll
    for (int i = 0; i < 8; ++i) {
        const int j = g + 32 * i;       // block index 0..255
        hs += part[j * 16 + e];
        cs += part[j * 16 + 8 + e];
    }
    lh[t] = hs; lc[t] = cs;
    __syncthreads();
    if (t < 8) {
        float H = 0.f, C = 0.f;
        for (int gg = 0; gg < 32; ++gg) { H += lh[gg * 8 + t]; C += lc[gg * 8 + t]; }
        lt[t] = H * C;
    }
    __syncthreads();
    if (t == 0) {
        float s = 0.f;
        for (int e2 = 0; e2 < E_N; ++e2) s += lt[e2];
        *out_loss = (float)E_N * s / ((float)B_N * (float)B_N);
    }
}

extern "C" void kernel_launch(void* const* d_in, const int* in_sizes, int n_in,
                              void* d_out, int out_size, void* d_ws, size_t ws_size,
                              hipStream_t stream) {
    (void)in_sizes; (void)n_in; (void)out_size; (void)ws_size;
    const float* x     = (const float*)d_in[0];
    const float* noise = (const float*)d_in[1];
    const float* gw    = (const float*)d_in[2];
    const float* gb    = (const float*)d_in[3];
    const float* ew    = (const float*)d_in[4];
    const float* eb    = (const float*)d_in[5];

    float* out      = (float*)d_out;            // [B,2]
    float* mask_out = out + (size_t)B_N * 2;    // [B,E]
    float* loss     = mask_out + (size_t)B_N * E_N;  // scalar
    float* ws       = (float*)d_ws;

    moe_prep_kernel<<<1, 256, 0, stream>>>(gw, gb, ew, eb, ws);
    moe_main_kernel<<<B_N / 16, 512, 0, stream>>>(x, noise, ws, ws + WS_PART, out, mask_out);
    moe_loss_kernel<<<1, 256, 0, stream>>>(ws + WS_PART, loss);
}